// AttentionTGN_47562467836659
// MI455X (gfx1250) — compile-verified
//
#include <hip/hip_runtime.h>
#include <hip/hip_bf16.h>
#include <math.h>

typedef __attribute__((ext_vector_type(2))) float v2f;
typedef __attribute__((ext_vector_type(8))) float v8f;

#define MTILE   32    // rows per block (2 x 16-row WMMA subtiles)
#define PSTRIDE 132   // 32x128 prev tile, padded: (132 % 64 == 4) -> conflict-free row reads
#define ASTRIDE 516   // 32x512 aggr tile, padded: (516 % 64 == 4) -> conflict-free row reads
                      // row strides: 528 B / 2064 B -> both 16B multiples (B128-friendly)

// ---------------------------------------------------------------------------
// Kernel 1: init outputs/workspace. assoc_out = assoc_in (zeros), lu_ws = 0,
// poskey = 0 (0 == "no message for this node").
// ---------------------------------------------------------------------------
__global__ void tgn_init(const int* __restrict__ assoc_in, int* __restrict__ assoc_out,
                         int* __restrict__ lu_ws, unsigned long long* __restrict__ poskey,
                         int num_nodes, int B) {
    int i = blockIdx.x * blockDim.x + threadIdx.x;
    if (i < num_nodes) { assoc_out[i] = assoc_in[i]; lu_ws[i] = 0; }
    if (i < B)         { poskey[i] = 0ULL; }
}

// ---------------------------------------------------------------------------
// Kernel 2: assoc[n_id[i]] = i
// ---------------------------------------------------------------------------
__global__ void tgn_scatter_assoc(const int* __restrict__ n_id, int* __restrict__ assoc_out, int B) {
    int i = blockIdx.x * blockDim.x + threadIdx.x;
    if (i < B) assoc_out[n_id[i]] = i;
}

// ---------------------------------------------------------------------------
// Kernel 3: per event e in [0,2E):  (first E = "s" set, second E = "d" set)
//   - lu_ws[node] = max(lu_ws[node], t+1)                 (segment_max over 1M nodes)
//   - poskey[assoc[node]] = max(key), key = (t<<32)|(e+1) (LastAggregator: max t, tie -> max e)
// ---------------------------------------------------------------------------
__global__ void tgn_events(const int* __restrict__ src_s, const int* __restrict__ t_s,
                           const int* __restrict__ src_d, const int* __restrict__ t_d,
                           const int* __restrict__ assoc_out,
                           unsigned long long* __restrict__ poskey,
                           int* __restrict__ lu_ws, int E) {
    int e = blockIdx.x * blockDim.x + threadIdx.x;
    if (e >= 2 * E) return;
    bool isS = e < E;
    int  er  = isS ? e : e - E;
    int  node = isS ? src_s[er] : src_d[er];
    int  t    = isS ? t_s[er]   : t_d[er];
    atomicMax(&lu_ws[node], t + 1);
    int seg = assoc_out[node];
    unsigned long long key = ((unsigned long long)(unsigned)t << 32) | (unsigned)(e + 1);
    atomicMax(&poskey[seg], key);
}

// ---------------------------------------------------------------------------
// Kernel 4: updated_last_update[i] = cnt>0 ? max_t : 0  ==  max(lu_ws[n_id]-1, 0)
// ---------------------------------------------------------------------------
__global__ void tgn_last_update(const int* __restrict__ n_id, const int* __restrict__ lu_ws,
                                int* __restrict__ out_lu, int B) {
    int i = blockIdx.x * blockDim.x + threadIdx.x;
    if (i < B) {
        int v = lu_ws[n_id[i]] - 1;
        out_lu[i] = v > 0 ? v : 0;
    }
}

// ---------------------------------------------------------------------------
// Kernel 5 (main): fused gather + 3 GEMMs + gated update, fp32 WMMA 16x16x4.
// One block = 32 node-rows (two 16-row WMMA subtiles). 8 waves, each owns a
// 16-wide N-tile of the 128 output cols and carries two accumulator sets, so
// every weight (B-operand) fetch feeds two WMMAs. The k- and v-GEMMs share
// the aggr A-operand, so they run in one fused K=512 loop (one ds_load feeds
// four WMMAs). aggr rows (incl. cos time-encoding) are gathered straight
// into LDS with branch-free B128 bursts — the [B,512] matrix never touches HBM.
// ---------------------------------------------------------------------------
__global__ __launch_bounds__(256)
void tgn_main(const unsigned long long* __restrict__ poskey,
              const float* __restrict__ nid_prev,
              const float* __restrict__ src_prev,
              const float* __restrict__ dst_prev,
              const float* __restrict__ raw_s,
              const float* __restrict__ raw_d,
              const int*   __restrict__ last_update,
              const float* __restrict__ time_w,
              const float* __restrict__ time_b,
              const float* __restrict__ Wq,
              const float* __restrict__ Wk,
              const float* __restrict__ Wv,
              float* __restrict__ out_mem,
              int E)
{
    __shared__ alignas(16) float ldsPrev[MTILE * PSTRIDE];   // 32 x 128 (padded)  ~16.9 KB
    __shared__ alignas(16) float ldsAggr[MTILE * ASTRIDE];   // 32 x 512 (padded)  ~66.0 KB
    __shared__ float ldsAcc[MTILE];                          // alpha dot-product accumulators
    __shared__ float ldsSig[MTILE];                          // sigmoid(alpha)

    const int tid = threadIdx.x;
    const int b0  = blockIdx.x * MTILE;

    // ---- cooperative tile load / gather: 8 threads per row, B128 bursts ----
    {
        const int r  = tid >> 3;   // row 0..31
        const int cg = tid & 7;    // float4-phase 0..7 (8 lanes x 16B = 128B bursts)
        const int node = b0 + r;

        // prev tile: 32 float4 per row
        const float4* pv = (const float4*)(nid_prev + (size_t)node * 128);
        float4* lp4 = (float4*)&ldsPrev[r * PSTRIDE];
        #pragma unroll
        for (int j = 0; j < 4; ++j) lp4[cg + 8 * j] = pv[cg + 8 * j];

        // aggr tile: 128 float4 per row in 4 segments of 32
        float4* la4 = (float4*)&ldsAggr[r * ASTRIDE];
        unsigned long long key = poskey[node];
        if (key == 0ULL) {
            float4 z = make_float4(0.f, 0.f, 0.f, 0.f);
            #pragma unroll
            for (int j = 0; j < 16; ++j) la4[cg + 8 * j] = z;
        } else {
            int  e   = (int)(unsigned)(key & 0xFFFFFFFFULL) - 1;
            int  tt  = (int)(key >> 32);
            bool isS = e < E;
            int  er  = isS ? e : e - E;
            const float* raw = isS ? raw_s : raw_d;
            float rel_t = (float)(tt - last_update[er]);
            size_t base = (size_t)er * 128;
            const float4* s4 = (const float4*)(src_prev + base);
            const float4* d4 = (const float4*)(dst_prev + base);
            const float4* r4 = (const float4*)(raw + base);
            const float4* tw4 = (const float4*)time_w;
            const float4* tb4 = (const float4*)time_b;
            #pragma unroll
            for (int j = 0; j < 4; ++j) { int f = cg + 8 * j; la4[f]      = s4[f]; }
            #pragma unroll
            for (int j = 0; j < 4; ++j) { int f = cg + 8 * j; la4[32 + f] = d4[f]; }
            #pragma unroll
            for (int j = 0; j < 4; ++j) { int f = cg + 8 * j; la4[64 + f] = r4[f]; }
            #pragma unroll
            for (int j = 0; j < 4; ++j) {
                int f = cg + 8 * j;
                float4 tw = tw4[f], tb = tb4[f], o;
                o.x = cosf(rel_t * tw.x + tb.x);
                o.y = cosf(rel_t * tw.y + tb.y);
                o.z = cosf(rel_t * tw.z + tb.z);
                o.w = cosf(rel_t * tw.w + tb.w);
                la4[96 + f] = o;
            }
        }
        if (tid < MTILE) ldsAcc[tid] = 0.0f;
    }
    __syncthreads();

    const int lane = tid & 31;
    const int wid  = tid >> 5;
    const int n0   = wid * 16;      // this wave's N-tile
    const int ln   = lane & 15;     // M (for A) / N (for B,D)
    const int lh   = lane >> 4;     // half: 0 -> K=k..k+1, 1 -> K=k+2..k+3
    const int koff = lh * 2;

    // ---- q = prev @ Wq  (K = 128), two 16-row subtiles per wave ----
    v8f qacc0 = {}, qacc1 = {};
    #pragma unroll 8
    for (int k = 0; k < 128; k += 4) {
        v2f a0, a1, b;
        const float* ap = &ldsPrev[ln * PSTRIDE + k + koff];
        a0.x = ap[0];              a0.y = ap[1];
        a1.x = ap[16 * PSTRIDE];   a1.y = ap[16 * PSTRIDE + 1];
        const float* bp = &Wq[(k + koff) * 128 + n0 + ln];
        b.x = bp[0]; b.y = bp[128];
        qacc0 = __builtin_amdgcn_wmma_f32_16x16x4_f32(false, a0, false, b, (short)0, qacc0, false, false);
        qacc1 = __builtin_amdgcn_wmma_f32_16x16x4_f32(false, a1, false, b, (short)0, qacc1, false, false);
    }

    // ---- fused k = aggr @ Wk and v = aggr @ Wv  (K = 512, shared A) ----
    v8f kacc0 = {}, kacc1 = {}, vacc0 = {}, vacc1 = {};
    #pragma unroll 4
    for (int k = 0; k < 512; k += 4) {
        v2f a0, a1, bk, bv;
        const float* ap = &ldsAggr[ln * ASTRIDE + k + koff];
        a0.x = ap[0];              a0.y = ap[1];
        a1.x = ap[16 * ASTRIDE];   a1.y = ap[16 * ASTRIDE + 1];
        const int woff = (k + koff) * 128 + n0 + ln;
        const float* bkp = &Wk[woff];
        const float* bvp = &Wv[woff];
        bk.x = bkp[0]; bk.y = bkp[128];
        bv.x = bvp[0]; bv.y = bvp[128];
        kacc0 = __builtin_amdgcn_wmma_f32_16x16x4_f32(false, a0, false, bk, (short)0, kacc0, false, false);
        kacc1 = __builtin_amdgcn_wmma_f32_16x16x4_f32(false, a1, false, bk, (short)0, kacc1, false, false);
        vacc0 = __builtin_amdgcn_wmma_f32_16x16x4_f32(false, a0, false, bv, (short)0, vacc0, false, false);
        vacc1 = __builtin_amdgcn_wmma_f32_16x16x4_f32(false, a1, false, bv, (short)0, vacc1, false, false);
    }

    // ---- alpha = sigmoid( sum_n q*k / sqrt(128) ) ----
    // D layout: lane holds col N=ln, VGPR i holds row (i + 8*lh) of the subtile.
    #pragma unroll
    for (int i = 0; i < 8; ++i) {
        float p0 = qacc0[i] * kacc0[i];
        float p1 = qacc1[i] * kacc1[i];
        p0 += __shfl_xor(p0, 1, 32);
        p1 += __shfl_xor(p1, 1, 32);
        p0 += __shfl_xor(p0, 2, 32);
        p1 += __shfl_xor(p1, 2, 32);
        p0 += __shfl_xor(p0, 4, 32);
        p1 += __shfl_xor(p1, 4, 32);
        p0 += __shfl_xor(p0, 8, 32);         // reduced within each 16-lane half
        p1 += __shfl_xor(p1, 8, 32);
        if (ln == 0) {
            atomicAdd(&ldsAcc[i + 8 * lh], p0);
            atomicAdd(&ldsAcc[16 + i + 8 * lh], p1);
        }
    }
    __syncthreads();
    if (tid < MTILE) {
        float x = ldsAcc[tid] * 0.08838834764831845f;   // 1/sqrt(128)
        ldsSig[tid] = 1.0f / (1.0f + expf(-x));
    }
    __syncthreads();

    // ---- updated_memory = prev + alpha * (v - prev) ----
    #pragma unroll
    for (int i = 0; i < 8; ++i) {
        int r0 = i + 8 * lh;
        int r1 = 16 + r0;
        float p0 = ldsPrev[r0 * PSTRIDE + n0 + ln];
        float p1 = ldsPrev[r1 * PSTRIDE + n0 + ln];
        out_mem[(size_t)(b0 + r0) * 128 + n0 + ln] = p0 + ldsSig[r0] * (vacc0[i] - p0);
        out_mem[(size_t)(b0 + r1) * 128 + n0 + ln] = p1 + ldsSig[r1] * (vacc1[i] - p1);
    }
}

// ---------------------------------------------------------------------------
// Launcher
// ---------------------------------------------------------------------------
extern "C" void kernel_launch(void* const* d_in, const int* in_sizes, int n_in,
                              void* d_out, int out_size, void* d_ws, size_t ws_size,
                              hipStream_t stream) {
    const int*   n_id        = (const int*)  d_in[0];
    const float* nid_prev    = (const float*)d_in[1];
    const int*   src_s       = (const int*)  d_in[2];
    // d_in[3] = dst_s  (unused by the reference math)
    const int*   t_s         = (const int*)  d_in[4];
    const float* raw_s       = (const float*)d_in[5];
    const int*   src_d       = (const int*)  d_in[6];
    // d_in[7] = dst_d  (unused)
    const int*   t_d         = (const int*)  d_in[8];
    const float* raw_d       = (const float*)d_in[9];
    const float* src_prev    = (const float*)d_in[10];
    const float* dst_prev    = (const float*)d_in[11];
    const int*   last_update = (const int*)  d_in[12];
    const int*   assoc_in    = (const int*)  d_in[13];
    const float* time_w      = (const float*)d_in[14];
    const float* time_b      = (const float*)d_in[15];
    const float* Wq          = (const float*)d_in[16];
    const float* Wk          = (const float*)d_in[17];
    const float* Wv          = (const float*)d_in[18];

    const int B         = in_sizes[0];
    const int E         = in_sizes[2];
    const int NUM_NODES = in_sizes[13];

    // d_out layout: [B*128] f32 updated_memory | [B] i32 last_update | [NUM_NODES] i32 assoc
    float* out_mem   = (float*)d_out;
    int*   out_lu    = (int*)((float*)d_out + (size_t)B * 128);
    int*   out_assoc = out_lu + B;

    // workspace: [B] u64 poskey | [NUM_NODES] i32 lu_ws   (~5 MB)
    unsigned long long* poskey = (unsigned long long*)d_ws;
    int* lu_ws = (int*)((char*)d_ws + (size_t)B * 8);

    const int T = 256;
    const int mx = NUM_NODES > B ? NUM_NODES : B;

    tgn_init<<<(mx + T - 1) / T, T, 0, stream>>>(assoc_in, out_assoc, lu_ws, poskey, NUM_NODES, B);
    tgn_scatter_assoc<<<(B + T - 1) / T, T, 0, stream>>>(n_id, out_assoc, B);
    tgn_events<<<(2 * E + T - 1) / T, T, 0, stream>>>(src_s, t_s, src_d, t_d,
                                                      out_assoc, poskey, lu_ws, E);
    tgn_last_update<<<(B + T - 1) / T, T, 0, stream>>>(n_id, lu_ws, out_lu, B);
    tgn_main<<<B / MTILE, 256, 0, stream>>>(poskey, nid_prev, src_prev, dst_prev,
                                            raw_s, raw_d, last_update, time_w, time_b,
                                            Wq, Wk, Wv, out_mem, E);
}